// HGT_26688926777649
// MI455X (gfx1250) — compile-verified
//
#include <hip/hip_runtime.h>
#include <hip/hip_bf16.h>
#include <math.h>

// ---------------- problem constants (match reference) ----------------
#define HGT_H    2
#define HGT_D    32
#define HGT_HID  64
#define HGT_L    2
#define HGT_OUT  16
#define HGT_NP   50000
#define HGT_NF   50000
#define HGT_E    1000000
#define HGT_IN   32
#define HGT_G    256

typedef float v2f __attribute__((ext_vector_type(2)));
typedef float v8f __attribute__((ext_vector_type(8)));

__device__ __forceinline__ float gelu_exact(float x) {
  return 0.5f * x * (1.0f + erff(x * 0.70710678118654752440f));
}

__device__ __forceinline__ void atomicMaxF(float* addr, float v) {
  // order-preserving: positives via signed int max, negatives via unsigned min
  if (v >= 0.0f) atomicMax((int*)addr, __float_as_int(v));
  else           atomicMin((unsigned int*)addr, __float_as_uint(v));
}

// ---------------- generic fill ----------------
__global__ void fill_kernel(float* __restrict__ p, float v, int n) {
  int i = blockIdx.x * blockDim.x + threadIdx.x;
  if (i < n) p[i] = v;
}

// ---------------- fused relation weights ----------------
// Wf[c, h*32+e'] = sum_d W[c, h*32+d] * rel[h, d, e'];  bf likewise from bvec.
__global__ void fuse_rel_kernel(const float* __restrict__ W,
                                const float* __restrict__ bvec,
                                const float* __restrict__ rel,
                                float* __restrict__ Wf, float* __restrict__ bf) {
  int tid = blockIdx.x * blockDim.x + threadIdx.x;  // 0..4095
  if (tid >= HGT_HID * HGT_HID) return;
  int c = tid >> 6;
  int o = tid & 63;
  int h = o >> 5;
  int ep = o & 31;
  const float* relh = rel + h * HGT_D * HGT_D;  // [d][e']
  float s = 0.0f, sb = 0.0f;
  #pragma unroll 8
  for (int d = 0; d < HGT_D; ++d) {
    float r = relh[d * HGT_D + ep];
    s += W[c * HGT_HID + h * HGT_D + d] * r;
    if (c == 0) sb += bvec[h * HGT_D + d] * r;
  }
  Wf[c * HGT_HID + o] = s;
  if (c == 0) bf[o] = sb;
}

// ---------------- WMMA fp32 GEMM: C[M,64] = epi(actA(A[M,K]) @ W[K,64] + bias) ----------------
// MODEA: 0 = identity, 1 = exact GELU on A load (compile-time specialized)
// MODEC: 0 = ReLU, 1 = none, 2 = skip blend: a*x + (1-a)*prev, a = sigmoid(*skipv)
// One wave -> 16 rows x 64 cols via four v_wmma_f32_16x16x4_f32 accumulators.
// B (weights, 16 KB max) is staged once per block into LDS as interleaved
// float2 {W[2k][n], W[2k+1][n]} so each per-lane fragment pair is one ds_load_b64.
template <int KVAL, int MODEA, int MODEC>
__global__ void __launch_bounds__(256)
gemm_n64_wmma(const float* __restrict__ A, const float* __restrict__ W,
              const float* __restrict__ bias, float* __restrict__ C,
              const float* __restrict__ prev, const float* __restrict__ skipv,
              int M) {
  __shared__ float2 WB[(KVAL / 2) * HGT_HID];     // K=64 -> 16 KB of the 320 KB WGP LDS

  // cooperative stage of W into LDS (all waves participate, then barrier,
  // THEN the wave-uniform early-out -> no divergent-barrier hazard)
  for (int i = threadIdx.x; i < (KVAL / 2) * HGT_HID; i += 256) {
    int kp = i >> 6;
    int n  = i & 63;
    WB[i] = make_float2(W[(size_t)(2 * kp) * HGT_HID + n],
                        W[(size_t)(2 * kp + 1) * HGT_HID + n]);
  }
  __syncthreads();

  const int lane = threadIdx.x & 31;
  const int wave = threadIdx.x >> 5;
  const int rowTile = blockIdx.x * 8 + wave;      // wave-uniform
  if (rowTile * 16 >= M) return;                  // EXEC stays all-ones per wave
  const int m0 = rowTile * 16;
  const int lm = lane & 15;                       // M (for A) / N (for B,C)
  const int hi = lane >> 4;                       // lane-half selects K pair / M+8

  v8f acc[4] = {};                                // 4 x (16x16) column tiles
  #pragma unroll
  for (int k0 = 0; k0 < KVAL; k0 += 4) {
    // A fragment: lane holds (m0+lm, k0+2*hi .. +1) -> one global_load_b64
    const float2 av = *(const float2*)(A + (size_t)(m0 + lm) * KVAL + (k0 + 2 * hi));
    float a0 = av.x, a1 = av.y;
    if (MODEA) { a0 = gelu_exact(a0); a1 = gelu_exact(a1); }
    v2f af = {a0, a1};
    const int kp = (k0 >> 1) + hi;                // LDS row of interleaved K pair
    #pragma unroll
    for (int nt = 0; nt < 4; ++nt) {
      float2 bv = WB[kp * HGT_HID + nt * 16 + lm];  // ds_load_b64
      v2f bf = {bv.x, bv.y};
      acc[nt] = __builtin_amdgcn_wmma_f32_16x16x4_f32(
          false, af, false, bf, (short)0, acc[nt], false, false);
    }
  }

  float ab = 0.0f;
  if (MODEC == 2) ab = 1.0f / (1.0f + expf(-skipv[0]));
  #pragma unroll
  for (int nt = 0; nt < 4; ++nt) {
    const int col = nt * 16 + lm;
    const float b = bias[col];
    #pragma unroll
    for (int r = 0; r < 8; ++r) {
      const int row = m0 + r + 8 * hi;            // C layout: VGPR r -> M = r + 8*hi
      float x = acc[nt][r] + b;
      if (MODEC == 0)      x = fmaxf(x, 0.0f);
      else if (MODEC == 2) x = ab * x + (1.0f - ab) * prev[(size_t)row * HGT_HID + col];
      C[(size_t)row * HGT_HID + col] = x;
    }
  }
}

// ---------------- edge pass A: logits + segment max ----------------
// 16 lanes per (edge, head): coalesced 128B row reads + shfl_xor tree reduce.
__global__ void edge_logits_kernel(const float* __restrict__ q,
                                   const float* __restrict__ kr,
                                   const int* __restrict__ src,
                                   const int* __restrict__ dst,
                                   const float* __restrict__ prel, float pscale,
                                   float* __restrict__ elog, float* __restrict__ m) {
  int tid = blockIdx.x * blockDim.x + threadIdx.x;
  int pair = tid >> 4;                 // (edge, head)
  int l16  = tid & 15;
  int ei = pair >> 1, h = pair & 1;
  if (ei >= HGT_E) return;
  int sN = src[ei], dN = dst[ei];
  const float2* qp = (const float2*)(q  + (size_t)dN * HGT_HID + h * HGT_D);
  const float2* kp = (const float2*)(kr + (size_t)sN * HGT_HID + h * HGT_D);
  float2 a = qp[l16];
  float2 b = kp[l16];
  float v = a.x * b.x + a.y * b.y;
  v += __shfl_xor(v, 8);               // xor masks stay inside the 16-lane group
  v += __shfl_xor(v, 4);
  v += __shfl_xor(v, 2);
  v += __shfl_xor(v, 1);
  if (l16 == 0) {
    float lg = v * prel[h] * pscale;
    elog[pair] = lg;
    atomicMaxF(&m[dN * 2 + h], lg);
  }
}

// ---------------- edge pass B: exp + segment sum ----------------
__global__ void edge_expsum_kernel(float* __restrict__ elog,
                                   const float* __restrict__ m,
                                   float* __restrict__ s,
                                   const int* __restrict__ dst) {
  int tid = blockIdx.x * blockDim.x + threadIdx.x;
  int ei = tid >> 1, h = tid & 1;
  if (ei >= HGT_E) return;
  int dN = dst[ei];
  float v = expf(elog[tid] - m[dN * 2 + h]);
  elog[tid] = v;
  atomicAdd(&s[dN * 2 + h], v);
}

// ---------------- edge pass C: alpha * v_rel scatter-add ----------------
__global__ void edge_aggregate_kernel(const float* __restrict__ elog,
                                      const float* __restrict__ s,
                                      const float* __restrict__ vrel,
                                      float* __restrict__ agg,
                                      const int* __restrict__ src,
                                      const int* __restrict__ dst) {
  int tid = blockIdx.x * blockDim.x + threadIdx.x;  // E*64 threads
  int ei = tid >> 6, c = tid & 63;
  if (ei >= HGT_E) return;
  int sN = src[ei], dN = dst[ei];
  int h = c >> 5;
  float alpha = elog[ei * 2 + h] / (s[dN * 2 + h] + 1e-16f);
  atomicAdd(&agg[(size_t)dN * HGT_HID + c], alpha * vrel[(size_t)sN * HGT_HID + c]);
}

// ---------------- pooling ----------------
__global__ void pool_sum_kernel(const float* __restrict__ xs,
                                const int* __restrict__ bidx,
                                float* __restrict__ ps, float* __restrict__ cnt,
                                int Nn) {
  int tid = blockIdx.x * blockDim.x + threadIdx.x;  // Nn*64
  int n = tid >> 6, c = tid & 63;
  if (n >= Nn) return;
  int g = bidx[n];
  atomicAdd(&ps[(size_t)g * HGT_HID + c], xs[(size_t)n * HGT_HID + c]);
  if (c == 0) atomicAdd(&cnt[g], 1.0f);
}

__global__ void pooled_out_kernel(const float* __restrict__ ps0, const float* __restrict__ cnt0,
                                  const float* __restrict__ ps1, const float* __restrict__ cnt1,
                                  const float* __restrict__ outW, const float* __restrict__ outb,
                                  float* __restrict__ out) {
  int tid = blockIdx.x * blockDim.x + threadIdx.x;  // G*16
  int g = tid >> 4, o = tid & 15;
  if (g >= HGT_G) return;
  float c0 = fmaxf(cnt0[g], 1.0f);
  float c1 = fmaxf(cnt1[g], 1.0f);
  float acc = outb[o];
  for (int c = 0; c < HGT_HID; ++c) {
    float hg = ps0[g * HGT_HID + c] / c0 + ps1[g * HGT_HID + c] / c1;
    acc += hg * outW[c * HGT_OUT + o];
  }
  out[(size_t)g * HGT_OUT + o] = acc;
}

// ---------------- host orchestration ----------------
static inline int cdiv(long long a, long long b) { return (int)((a + b - 1) / b); }

extern "C" void kernel_launch(void* const* d_in, const int* in_sizes, int n_in,
                              void* d_out, int out_size, void* d_ws, size_t ws_size,
                              hipStream_t stream) {
  (void)in_sizes; (void)n_in; (void)out_size; (void)ws_size;
  const float* x_p   = (const float*)d_in[0];
  const float* x_f   = (const float*)d_in[1];
  const int*   ei_pf = (const int*)d_in[2];
  const int*   ei_fp = (const int*)d_in[3];
  const int*   b_p   = (const int*)d_in[4];
  const int*   b_f   = (const int*)d_in[5];
  const float* lin_W = (const float*)d_in[6];
  const float* lin_b = (const float*)d_in[7];
  const float* kW    = (const float*)d_in[8];
  const float* kb    = (const float*)d_in[9];
  const float* qW    = (const float*)d_in[10];
  const float* qb    = (const float*)d_in[11];
  const float* vW    = (const float*)d_in[12];
  const float* vb    = (const float*)d_in[13];
  const float* aW    = (const float*)d_in[14];
  const float* abv   = (const float*)d_in[15];
  const float* skip  = (const float*)d_in[16];
  const float* a_rel = (const float*)d_in[17];
  const float* m_rel = (const float*)d_in[18];
  const float* p_rel = (const float*)d_in[19];
  const float* out_W = (const float*)d_in[20];
  const float* out_b = (const float*)d_in[21];
  float* out = (float*)d_out;

  const int Ns[2] = {HGT_NP, HGT_NF};
  const int* eidx[2] = {ei_pf, ei_fp};
  const size_t U = (size_t)HGT_NP * HGT_HID;  // N_P == N_F

  float* w = (float*)d_ws;
  float* xs[2]  = {w + 0 * U, w + 1 * U};
  float* qb_[2] = {w + 2 * U, w + 3 * U};
  float* kr[2]  = {w + 4 * U, w + 5 * U};
  float* vr[2]  = {w + 6 * U, w + 7 * U};
  float* agg[2] = {w + 8 * U, w + 9 * U};
  float* elog   = w + 10 * U;                       // 2E floats
  float* mbuf   = elog + (size_t)2 * HGT_E;         // N*H
  float* sbuf   = mbuf + (size_t)HGT_NP * HGT_H;    // N*H
  float* kWf    = sbuf + (size_t)HGT_NP * HGT_H;    // L*2*64*64
  float* kbf    = kWf + HGT_L * 2 * HGT_HID * HGT_HID;  // L*2*64
  float* vWf    = kbf + HGT_L * 2 * HGT_HID;
  float* vbf    = vWf + HGT_L * 2 * HGT_HID * HGT_HID;
  float* ps0    = vbf + HGT_L * 2 * HGT_HID;        // G*64
  float* ps1    = ps0 + HGT_G * HGT_HID;
  float* cnt0   = ps1 + HGT_G * HGT_HID;            // G
  float* cnt1   = cnt0 + HGT_G;

  const int TB = 256;
  const float pscale = 1.0f / sqrtf((float)HGT_D);
  const int gemmGrid = cdiv(HGT_NP / 16, 8);        // 3125 tiles / 8 waves

  // 1) fused relation weights: kW_f = kW . blockdiag(a_rel); vW_f = vW . blockdiag(m_rel)
  for (int l = 0; l < HGT_L; ++l) {
    for (int e = 0; e < 2; ++e) {                   // src type == e
      const int sl = l * 2 + e;
      hipLaunchKernelGGL(fuse_rel_kernel, dim3(16), dim3(TB), 0, stream,
          kW + (size_t)sl * HGT_HID * HGT_HID, kb + (size_t)sl * HGT_HID,
          a_rel + (size_t)sl * HGT_H * HGT_D * HGT_D,
          kWf + (size_t)sl * HGT_HID * HGT_HID, kbf + (size_t)sl * HGT_HID);
      hipLaunchKernelGGL(fuse_rel_kernel, dim3(16), dim3(TB), 0, stream,
          vW + (size_t)sl * HGT_HID * HGT_HID, vb + (size_t)sl * HGT_HID,
          m_rel + (size_t)sl * HGT_H * HGT_D * HGT_D,
          vWf + (size_t)sl * HGT_HID * HGT_HID, vbf + (size_t)sl * HGT_HID);
    }
  }

  // 2) input linear + ReLU (K=32)
  hipLaunchKernelGGL(HIP_KERNEL_NAME(gemm_n64_wmma<HGT_IN, 0, 0>),
      dim3(gemmGrid), dim3(TB), 0, stream,
      x_p, lin_W + 0, lin_b + 0, xs[0], nullptr, nullptr, HGT_NP);
  hipLaunchKernelGGL(HIP_KERNEL_NAME(gemm_n64_wmma<HGT_IN, 0, 0>),
      dim3(gemmGrid), dim3(TB), 0, stream,
      x_f, lin_W + HGT_IN * HGT_HID, lin_b + HGT_HID, xs[1], nullptr, nullptr, HGT_NF);

  // 3) layers
  for (int l = 0; l < HGT_L; ++l) {
    for (int t = 0; t < 2; ++t) {
      const int sl = l * 2 + t;
      // q (plain), k_rel and v_rel via fused weights — three WMMA GEMMs
      hipLaunchKernelGGL(HIP_KERNEL_NAME(gemm_n64_wmma<HGT_HID, 0, 1>),
          dim3(gemmGrid), dim3(TB), 0, stream,
          xs[t], qW + (size_t)sl * HGT_HID * HGT_HID, qb + (size_t)sl * HGT_HID,
          qb_[t], nullptr, nullptr, Ns[t]);
      hipLaunchKernelGGL(HIP_KERNEL_NAME(gemm_n64_wmma<HGT_HID, 0, 1>),
          dim3(gemmGrid), dim3(TB), 0, stream,
          xs[t], kWf + (size_t)sl * HGT_HID * HGT_HID, kbf + (size_t)sl * HGT_HID,
          kr[t], nullptr, nullptr, Ns[t]);
      hipLaunchKernelGGL(HIP_KERNEL_NAME(gemm_n64_wmma<HGT_HID, 0, 1>),
          dim3(gemmGrid), dim3(TB), 0, stream,
          xs[t], vWf + (size_t)sl * HGT_HID * HGT_HID, vbf + (size_t)sl * HGT_HID,
          vr[t], nullptr, nullptr, Ns[t]);
      hipLaunchKernelGGL(fill_kernel, dim3(cdiv(U, TB)), dim3(TB), 0, stream,
          agg[t], 0.0f, (int)U);
    }
    for (int e = 0; e < 2; ++e) {
      const int st = e, dt = 1 - e;
      const int* src = eidx[e];
      const int* dst = eidx[e] + HGT_E;
      hipLaunchKernelGGL(fill_kernel, dim3(cdiv((long long)HGT_NP * HGT_H, TB)), dim3(TB), 0, stream,
          mbuf, -3.402823466e38f, HGT_NP * HGT_H);
      hipLaunchKernelGGL(fill_kernel, dim3(cdiv((long long)HGT_NP * HGT_H, TB)), dim3(TB), 0, stream,
          sbuf, 0.0f, HGT_NP * HGT_H);
      hipLaunchKernelGGL(edge_logits_kernel,
          dim3(cdiv((long long)HGT_E * HGT_H * 16, TB)), dim3(TB), 0, stream,
          qb_[dt], kr[st], src, dst, p_rel + (size_t)(l * 2 + e) * HGT_H, pscale, elog, mbuf);
      hipLaunchKernelGGL(edge_expsum_kernel,
          dim3(cdiv((long long)HGT_E * HGT_H, TB)), dim3(TB), 0, stream,
          elog, mbuf, sbuf, dst);
      hipLaunchKernelGGL(edge_aggregate_kernel,
          dim3(cdiv((long long)HGT_E * HGT_HID, TB)), dim3(TB), 0, stream,
          elog, sbuf, vr[st], agg[dt], src, dst);
    }
    for (int t = 0; t < 2; ++t) {
      const int sl = l * 2 + t;
      // xs = sigmoid(skip)*(gelu(agg)@aW + ab) + (1-sigmoid(skip))*xs   (GELU fused in A-load)
      hipLaunchKernelGGL(HIP_KERNEL_NAME(gemm_n64_wmma<HGT_HID, 1, 2>),
          dim3(gemmGrid), dim3(TB), 0, stream,
          agg[t], aW + (size_t)sl * HGT_HID * HGT_HID, abv + (size_t)sl * HGT_HID,
          xs[t], xs[t], skip + sl, Ns[t]);
    }
  }

  // 4) pooling + head
  hipLaunchKernelGGL(fill_kernel, dim3(cdiv(2 * HGT_G * HGT_HID + 2 * HGT_G, TB)), dim3(TB), 0, stream,
      ps0, 0.0f, 2 * HGT_G * HGT_HID + 2 * HGT_G);
  hipLaunchKernelGGL(pool_sum_kernel, dim3(cdiv((long long)HGT_NP * HGT_HID, TB)), dim3(TB), 0, stream,
      xs[0], b_p, ps0, cnt0, HGT_NP);
  hipLaunchKernelGGL(pool_sum_kernel, dim3(cdiv((long long)HGT_NF * HGT_HID, TB)), dim3(TB), 0, stream,
      xs[1], b_f, ps1, cnt1, HGT_NF);
  hipLaunchKernelGGL(pooled_out_kernel, dim3(cdiv(HGT_G * HGT_OUT, TB)), dim3(TB), 0, stream,
      ps0, cnt0, ps1, cnt1, out_W, out_b, out);
}